// Layer_MPNN_vertexPE_86878598463999
// MI455X (gfx1250) — compile-verified
//
#include <hip/hip_runtime.h>
#include <hip/hip_bf16.h>
#include <math.h>

// MI455X / gfx1250, wave32.
// B=16, d1=d2=dout=64, p1=p2=256. H is 256MB -> purely HBM-bound (~11us floor).
// Single pass over H producing both reductions; GEMMs via V_WMMA_F32_16X16X4_F32.

typedef __attribute__((ext_vector_type(2))) float v2f;
typedef __attribute__((ext_vector_type(4))) float v4f;
typedef __attribute__((ext_vector_type(8))) float v8f;

__device__ __forceinline__ float wave_sum(float v) {
  v += __shfl_xor(v, 16);
  v += __shfl_xor(v, 8);
  v += __shfl_xor(v, 4);
  v += __shfl_xor(v, 2);
  v += __shfl_xor(v, 1);
  return v;
}

// ---- K1: sums of A_AP / A_UE over last axis (wave per row) -----------------
__global__ void k_asum(const float* __restrict__ A_AP, const float* __restrict__ A_UE,
                       float* __restrict__ apSum, float* __restrict__ ueSum) {
  int wave = threadIdx.x >> 5, lane = threadIdx.x & 31;
  int r = blockIdx.x * 8 + wave;               // 2048 waves total
  const float* src; float* dst; int rr;
  if (r < 1024) { src = A_AP; dst = apSum; rr = r; }
  else          { src = A_UE; dst = ueSum; rr = r - 1024; }
  const float* p = src + (size_t)rr * 256;
  float s = 0.f;
#pragma unroll
  for (int i = 0; i < 8; ++i) s += p[i * 32 + lane];
  s = wave_sum(s);
  if (lane == 0) dst[rr] = s;
}

// ---- K2: bias[b][o] = (2/256) * sum_d Q2[o][d] * meanSum[b][d] -------------
__global__ void k_bias(const float* __restrict__ Q2_AP, const float* __restrict__ Q2_UE,
                       const float* __restrict__ apSum, const float* __restrict__ ueSum,
                       float* __restrict__ bias_ap, float* __restrict__ bias_ue) {
  int idx = blockIdx.x * 256 + threadIdx.x;    // 2048 threads
  int sel = idx >> 10, b = (idx >> 6) & 15, o = idx & 63;
  const float* Q2 = sel ? Q2_UE : Q2_AP;       // ap side pairs with ue_mean and v.v.
  const float* ms = sel ? apSum : ueSum;
  float s = 0.f;
#pragma unroll 8
  for (int d = 0; d < 64; ++d) s += Q2[o * 64 + d] * ms[b * 64 + d];
  (sel ? bias_ue : bias_ap)[b * 64 + o] = s * (2.0f / 256.0f);
}

// ---- K3: single-pass H reduction: row sums (over p2) + col sums (over p1) --
// One 256x256 f32 tile per block (blockIdx = b*64 + c). Deterministic: no
// float atomics; row halves are reduced via shfl tree, stored once each.
__global__ void k_hreduce(const float* __restrict__ H,
                          float* __restrict__ hsum_ue, float* __restrict__ hsum_ap) {
  __shared__ float srow[512];    // [row][half]
  __shared__ float scol[1024];   // [ty][col]
  int tid = threadIdx.x;
  int tx = tid & 63, ty = tid >> 6;
  int lane = tid & 31, wave = tid >> 5, h = wave & 1;
  const v4f* tp = (const v4f*)(H + (size_t)blockIdx.x * 65536);
  v4f ca = {0.f, 0.f, 0.f, 0.f};
  for (int it = 0; it < 64; ++it) {
    int r = it * 4 + ty;                       // uniform within wave
    v4f v = __builtin_nontemporal_load(&tp[r * 64 + tx]);  // streaming: bypass temporal caching
    ca.x += v.x; ca.y += v.y; ca.z += v.z; ca.w += v.w;
    float rs = wave_sum(v.x + v.y + v.z + v.w);
    if (lane == 0) srow[r * 2 + h] = rs;       // each slot written exactly once
  }
  ((v4f*)scol)[ty * 64 + tx] = ca;
  __syncthreads();
  size_t o = (size_t)blockIdx.x * 256 + tid;
  hsum_ue[o] = srow[tid * 2] + srow[tid * 2 + 1];                       // sum over p2
  hsum_ap[o] = scol[tid] + scol[256 + tid] + scol[512 + tid] + scol[768 + tid]; // sum over p1
}

// ---- K4: batched GEMMs with V_WMMA_F32_16X16X4_F32 -------------------------
// D = W(16x4) x X(4x16) + C.  A-frag: lanes0-15 K0/K1, lanes16-31 K2/K3.
// B-frag: VGPR0 = rows K0(l0-15)/K2(l16-31), VGPR1 = K1/K3.  C/D: M = v + 8*(lane>=16).
__device__ __forceinline__ void gemm_k64(const float* __restrict__ W,
                                         const float* __restrict__ X, float scl,
                                         int mln, int kg, int na, int nb,
                                         v8f& acc0, v8f& acc1) {
#pragma unroll 4
  for (int k4 = 0; k4 < 64; k4 += 4) {
    int kk = k4 + kg;                          // kg in {0,2}
    v2f w = *(const v2f*)(W + mln * 64 + kk);  // W[m][kk], W[m][kk+1]
    v2f av; av.x = w.x * scl; av.y = w.y * scl;
    v2f b0; b0.x = X[kk * 256 + na]; b0.y = X[(kk + 1) * 256 + na];
    v2f b1; b1.x = X[kk * 256 + nb]; b1.y = X[(kk + 1) * 256 + nb];
    acc0 = __builtin_amdgcn_wmma_f32_16x16x4_f32(false, av, false, b0, (short)0, acc0, false, false);
    acc1 = __builtin_amdgcn_wmma_f32_16x16x4_f32(false, av, false, b1, (short)0, acc1, false, false);
  }
}

__global__ void k_gemm(const float* __restrict__ Q1_AP, const float* __restrict__ Q1_UE,
                       const float* __restrict__ P1_AP, const float* __restrict__ P1_UE,
                       const float* __restrict__ A_AP, const float* __restrict__ A_UE,
                       const float* __restrict__ hsum_ue, const float* __restrict__ hsum_ap,
                       const float* __restrict__ bias_ap, const float* __restrict__ bias_ue,
                       float* __restrict__ out) {
  int bid = blockIdx.x;                        // 128 = nstrip(4) * b(16) * sel(2)
  int sel = bid & 1, b = (bid >> 1) & 15, ns = bid >> 5;
  int tid = threadIdx.x, wave = tid >> 5, lane = tid & 31;
  int ln = lane & 15, kg = (lane >> 4) * 2;
  int mt = wave & 3, np = wave >> 2;
  int M0 = mt * 16;
  int n0a = ns * 64 + np * 16, n0b = ns * 64 + (np + 2) * 16;
  const float* W1 = sel ? Q1_UE : Q1_AP;
  const float* W2 = sel ? P1_UE : P1_AP;
  const float* X1 = (sel ? A_UE : A_AP) + b * 16384;
  const float* X2 = (sel ? hsum_ap : hsum_ue) + b * 16384;
  const float* bias = (sel ? bias_ue : bias_ap) + b * 64;
  float* op = out + sel * 262144 + b * 16384;
  v8f acc0 = {}; v8f acc1 = {};
  int mln = M0 + ln, na = n0a + ln, nb = n0b + ln;
  gemm_k64(W1, X1, 2.0f,          mln, kg, na, nb, acc0, acc1); // 2*Q1@A
  gemm_k64(W2, X2, 0.1f / 256.0f, mln, kg, na, nb, acc0, acc1); // 0.1*P1@mean(H)
  int mbase = M0 + (lane >> 4) * 8;
#pragma unroll
  for (int v = 0; v < 8; ++v) {
    int m = mbase + v;
    float bi = bias[m];
    op[m * 256 + na] = fmaxf(acc0[v] + bi, 0.f);
    op[m * 256 + nb] = fmaxf(acc1[v] + bi, 0.f);
  }
}

// ---- K5: per-channel batch-norm statistics (training-mode, biased var) -----
__global__ void k_stats(const float* __restrict__ out, const float* __restrict__ gamma,
                        const float* __restrict__ beta, float* __restrict__ scsh) {
  __shared__ float ss[8], sq[8];
  int sel = blockIdx.x >> 6, o = blockIdx.x & 63;
  int tid = threadIdx.x, wave = tid >> 5, lane = tid & 31;
  const float* p = out + sel * 262144 + o * 256 + tid;
  float s = 0.f, q = 0.f;
#pragma unroll
  for (int b = 0; b < 16; ++b) {
    float v = p[b * 16384];
    s += v; q += v * v;
  }
  s = wave_sum(s); q = wave_sum(q);
  if (lane == 0) { ss[wave] = s; sq[wave] = q; }
  __syncthreads();
  if (tid == 0) {
    float S = 0.f, Q = 0.f;
    for (int i = 0; i < 8; ++i) { S += ss[i]; Q += sq[i]; }
    float mean = S * (1.0f / 4096.0f);
    float var  = Q * (1.0f / 4096.0f) - mean * mean;
    float sc = gamma[o] * rsqrtf(var + 1e-5f);
    scsh[(sel * 64 + o) * 2 + 0] = sc;
    scsh[(sel * 64 + o) * 2 + 1] = beta[o] - mean * sc;
  }
}

// ---- K6: apply BN affine in place ------------------------------------------
__global__ void k_bnapply(float* __restrict__ out, const float* __restrict__ scsh) {
  int fi = blockIdx.x * 256 + threadIdx.x;     // 131072 float4's
  int f0 = fi * 4;
  int o = (f0 >> 8) & 63;
  int sel = f0 >> 18;
  float sc = scsh[(sel * 64 + o) * 2 + 0];
  float sh = scsh[(sel * 64 + o) * 2 + 1];
  v4f v = ((v4f*)out)[fi];
  v.x = v.x * sc + sh; v.y = v.y * sc + sh;
  v.z = v.z * sc + sh; v.w = v.w * sc + sh;
  ((v4f*)out)[fi] = v;
}

extern "C" void kernel_launch(void* const* d_in, const int* in_sizes, int n_in,
                              void* d_out, int out_size, void* d_ws, size_t ws_size,
                              hipStream_t stream) {
  (void)in_sizes; (void)n_in; (void)out_size; (void)ws_size;
  const float* A_AP  = (const float*)d_in[0];
  const float* A_UE  = (const float*)d_in[1];
  const float* H     = (const float*)d_in[2];
  const float* Q1_AP = (const float*)d_in[3];
  const float* Q2_AP = (const float*)d_in[4];
  const float* Q1_UE = (const float*)d_in[5];
  const float* Q2_UE = (const float*)d_in[6];
  const float* P1_AP = (const float*)d_in[7];
  const float* P1_UE = (const float*)d_in[8];
  const float* gamma = (const float*)d_in[9];
  const float* beta  = (const float*)d_in[10];
  float* out = (float*)d_out;
  float* ws  = (float*)d_ws;

  float* apSum   = ws;                       // 1024
  float* ueSum   = ws + 1024;                // 1024
  float* bias_ap = ws + 2048;                // 1024
  float* bias_ue = ws + 3072;                // 1024
  float* hsum_ue = ws + 4096;                // 262144
  float* hsum_ap = ws + 4096 + 262144;       // 262144
  float* scsh    = ws + 4096 + 524288;       // 256

  k_asum   <<<256, 256, 0, stream>>>(A_AP, A_UE, apSum, ueSum);
  k_bias   <<<8,   256, 0, stream>>>(Q2_AP, Q2_UE, apSum, ueSum, bias_ap, bias_ue);
  k_hreduce<<<1024,256, 0, stream>>>(H, hsum_ue, hsum_ap);
  k_gemm   <<<128, 256, 0, stream>>>(Q1_AP, Q1_UE, P1_AP, P1_UE, A_AP, A_UE,
                                     hsum_ue, hsum_ap, bias_ap, bias_ue, out);
  k_stats  <<<128, 256, 0, stream>>>(out, gamma, beta, scsh);
  k_bnapply<<<512, 256, 0, stream>>>(out, scsh);
}